// FermiNet_3470333575739
// MI455X (gfx1250) — compile-verified
//
#include <hip/hip_runtime.h>
#include <hip/hip_bf16.h>
#include <math.h>

// ---- problem constants (match reference) ----
#define BW    8192   // walkers
#define NE    16     // electrons
#define NIONC 4
#define DHC   256
#define NDETC 16
#define NSPC  2
#define NSC   8      // electrons/orbitals per spin block
#define ASTR  258    // LDS row stride (floats), 258%64==2 -> conflict-free WMMA A reads

typedef __attribute__((ext_vector_type(2))) float v2f;
typedef __attribute__((ext_vector_type(8))) float v8f;

// 8x8 determinant, one column per lane across an 8-lane group.
// m[i] = A[row i][my column]. Gaussian elimination w/ partial pivoting.
// All 8 lanes of a group compute identical (sign, logdet).
__device__ __forceinline__ void det8(float m[8], float& sgn, float& logdet) {
    sgn = 1.0f;
    logdet = 0.0f;
#pragma unroll
    for (int k = 0; k < 8; ++k) {
        // broadcast column k to all lanes of the group
        float ck[8];
#pragma unroll
        for (int i = 0; i < 8; ++i) ck[i] = __shfl(m[i], k, 8);
        // partial pivot: argmax |A[i][k]| for i >= k (uniform across group)
        int p = k;
        float best = fabsf(ck[k]);
#pragma unroll
        for (int i = 0; i < 8; ++i) {
            if (i > k) {
                float v = fabsf(ck[i]);
                if (v > best) { best = v; p = i; }
            }
        }
        // conditional row swap k <-> p (compile-time register indices only)
#pragma unroll
        for (int i = 0; i < 8; ++i) {
            if (i > k) {
                bool c = (i == p);
                float tk = m[k],  ti = m[i];
                m[k]  = c ? ti : tk;   m[i]  = c ? tk : ti;
                float ak = ck[k], ai = ck[i];
                ck[k] = c ? ai : ak;   ck[i] = c ? ak : ai;
            }
        }
        if (p != k) sgn = -sgn;
        float piv = ck[k];
        if (piv < 0.0f) sgn = -sgn;
        logdet += logf(fabsf(piv));
        float inv = 1.0f / piv;
#pragma unroll
        for (int i = 0; i < 8; ++i) {
            if (i > k) {
                float f = ck[i] * inv;
                m[i] = fmaf(-f, m[k], m[i]);
            }
        }
    }
}

// One block = 256 threads = 8 waves. Block handles (spin s, walker-pair).
// Wave w handles det-pair w. Each wave's 16x16 WMMA tile holds 4 full 8x8
// orbital matrices -> solved in-register.
__global__ __launch_bounds__(256) void ferminet_main(
    const float* __restrict__ elec_pos,  // (B,16,3)
    const float* __restrict__ ion_pos,   // (4,3)
    const float* __restrict__ W1,        // (16,256)
    const float* __restrict__ b1,        // (256,)
    const float* __restrict__ Wlin,      // (16,2,256,8)
    const float* __restrict__ blin,      // (16,2,8)
    const float* __restrict__ decay,     // (16,2,4,8)
    const float* __restrict__ pim,       // (16,2,4,8)
    float* __restrict__ signs,           // (16*2, B)
    float* __restrict__ logs)            // (16*2, B)
{
    __shared__ float As[16 * ASTR];      // stream tile: 16 rows x 256 (padded)
    __shared__ float Dl[16 * NIONC];     // |r_e - R_i| per row

    const int t = threadIdx.x;
    const int s = blockIdx.x & 1;        // spin block
    const int wpair = blockIdx.x >> 1;   // walker pair

    // ---------- Phase 1: backflow features -> tanh MLP stream tile ----------
    {
        const int r  = t >> 4;             // tile row 0..15 = (walker-in-pair, e)
        const int cb = (t & 15) * 16;      // 16 consecutive output columns
        const int w  = wpair * 2 + (r >> 3);
        const int n  = s * NSC + (r & 7);
        const float ex = elec_pos[(w * NE + n) * 3 + 0];
        const float ey = elec_pos[(w * NE + n) * 3 + 1];
        const float ez = elec_pos[(w * NE + n) * 3 + 2];
        float feats[16];
#pragma unroll
        for (int i = 0; i < NIONC; ++i) {
            float dx = ex - ion_pos[i * 3 + 0];
            float dy = ey - ion_pos[i * 3 + 1];
            float dz = ez - ion_pos[i * 3 + 2];
            feats[i * 3 + 0] = dx;
            feats[i * 3 + 1] = dy;
            feats[i * 3 + 2] = dz;
            feats[12 + i] = sqrtf(dx * dx + dy * dy + dz * dz);
        }
        if ((t & 15) == 0) {
#pragma unroll
            for (int i = 0; i < NIONC; ++i) Dl[r * NIONC + i] = feats[12 + i];
        }
#pragma unroll
        for (int j = 0; j < 16; ++j) {
            const int col = cb + j;
            float acc = b1[col];
#pragma unroll
            for (int k = 0; k < 16; ++k)
                acc = fmaf(feats[k], W1[k * DHC + col], acc);
            As[r * ASTR + col] = tanhf(acc);
        }
    }
    __syncthreads();

    // ---------- Phase 2: phi tile via V_WMMA_F32_16X16X4_F32, K = 256 ----------
    const int lane  = t & 31;
    const int dpair = t >> 5;            // wave id = determinant pair
    const int mA    = lane & 15;         // A row (M)
    const int kh    = lane >> 4;         // K-half select
    const int ncol  = lane & 15;         // C/D column (N)
    const int dh    = ncol >> 3;         // det within pair
    const int o     = ncol & 7;          // orbital
    const int d     = dpair * 2 + dh;

    const float* Wp   = Wlin + ((size_t)(d * NSPC + s) * DHC) * NSC + o; // [h][o] @ Wp[h*8]
    const float* Arow = As + mA * ASTR + 2 * kh;

    v8f c = {};
#pragma unroll 4
    for (int kb = 0; kb < DHC; kb += 4) {
        v2f a, bb;
        a.x  = Arow[kb];
        a.y  = Arow[kb + 1];
        bb.x = Wp[(size_t)(kb + 2 * kh) * NSC];
        bb.y = Wp[(size_t)(kb + 2 * kh + 1) * NSC];
        c = __builtin_amdgcn_wmma_f32_16x16x4_f32(false, a, false, bb,
                                                  (short)0, c, false, false);
    }

    // ---------- Phase 3: bias + isotropic exponential envelope ----------
    const int wloc  = lane >> 4;             // walker within pair (= C lane-half)
    const int bwalk = wpair * 2 + wloc;
    const float bl = blin[(d * NSPC + s) * NSC + o];
    float dec[NIONC], piw[NIONC];
#pragma unroll
    for (int i = 0; i < NIONC; ++i) {
        const int idx = ((d * NSPC + s) * NIONC + i) * NSC + o;
        dec[i] = decay[idx];
        piw[i] = pim[idx];
    }
    float m8[8];   // column `o` of this lane's 8x8 orbital matrix, rows e=0..7
#pragma unroll
    for (int e = 0; e < 8; ++e) {
        float env = 0.0f;
#pragma unroll
        for (int i = 0; i < NIONC; ++i)
            env += piw[i] * expf(-dec[i] * Dl[(wloc * 8 + e) * NIONC + i]);
        m8[e] = (c[e] + bl) * env;
    }

    // ---------- Phase 4: in-register 8x8 slogdet per 8-lane group ----------
    float sgn, ld;
    det8(m8, sgn, ld);
    if ((lane & 7) == 0) {
        const size_t idx = (size_t)(d * NSPC + s) * BW + bwalk;
        signs[idx] = sgn;
        logs[idx]  = ld;
    }
}

// Per-walker: combine spins, sign-aware log-sum-exp over 16 determinants.
__global__ __launch_bounds__(256) void ferminet_reduce(
    const float* __restrict__ signs,   // (32, B)
    const float* __restrict__ logs,    // (32, B)
    float* __restrict__ out)           // (B,)
{
    const int b = blockIdx.x * blockDim.x + threadIdx.x;
    if (b >= BW) return;
    float sd[NDETC], ld[NDETC];
    float mx = -INFINITY;
#pragma unroll
    for (int d = 0; d < NDETC; ++d) {
        const float s0 = signs[(size_t)(d * 2 + 0) * BW + b];
        const float s1 = signs[(size_t)(d * 2 + 1) * BW + b];
        const float l  = logs[(size_t)(d * 2 + 0) * BW + b]
                       + logs[(size_t)(d * 2 + 1) * BW + b];
        sd[d] = s0 * s1;
        ld[d] = l;
        mx = fmaxf(mx, l);
    }
    float psi = 0.0f;
#pragma unroll
    for (int d = 0; d < NDETC; ++d)
        psi += sd[d] * expf(ld[d] - mx);
    out[b] = mx + logf(fabsf(psi));
}

extern "C" void kernel_launch(void* const* d_in, const int* in_sizes, int n_in,
                              void* d_out, int out_size, void* d_ws, size_t ws_size,
                              hipStream_t stream) {
    const float* elec_pos = (const float*)d_in[0];
    const float* ion_pos  = (const float*)d_in[1];
    const float* W1       = (const float*)d_in[2];
    const float* b1       = (const float*)d_in[3];
    const float* Wlin     = (const float*)d_in[4];
    const float* blin     = (const float*)d_in[5];
    const float* decay    = (const float*)d_in[6];
    const float* pim      = (const float*)d_in[7];

    float* signs = (float*)d_ws;                         // 16*2*8192 floats = 1 MB
    float* logs  = signs + (size_t)NDETC * NSPC * BW;    // +1 MB

    ferminet_main<<<dim3(NSPC * (BW / 2)), dim3(256), 0, stream>>>(
        elec_pos, ion_pos, W1, b1, Wlin, blin, decay, pim, signs, logs);

    ferminet_reduce<<<dim3(BW / 256), dim3(256), 0, stream>>>(
        signs, logs, (float*)d_out);
}